// WaveNet_11982958756771
// MI455X (gfx1250) — compile-verified
//
#include <hip/hip_runtime.h>
#include <hip/hip_bf16.h>
#include <math.h>

typedef float v2f __attribute__((ext_vector_type(2)));
typedef float v4f __attribute__((ext_vector_type(4)));
typedef float v8f __attribute__((ext_vector_type(8)));

// D = A(16x4,f32) * B(4x16,f32) + C(16x16,f32), full-fp32 WMMA on gfx1250.
#define WMMA_F32X4(a, b, c) \
  __builtin_amdgcn_wmma_f32_16x16x4_f32(false, (a), false, (b), (short)0, (c), false, false)

#define WN_B 8
#define WN_N 32768
#define WN_C 16
#define WN_L 30
#define WN_TPW 4                         // time tiles per wave in gated layers
#define WN_P2N (2 * WN_N)                // pair-interleaved channel-pair stride
#define WN_FRAG_PER_LAYER (8 * 32 * 2)   // 8 frags x 32 lanes x v2f

// h/agg live in workspace pair-interleaved: idx(c,t) = (c>>1)*2N + 2t + (c&1),
// so every (c, c+1) pair the WMMA fragments / epilogue need is one b64.

// Branch-free activations: use hw v_tanh_f32 if the builtin exists, else
// tanh(x) = 2*sigmoid(2x)-1 via v_exp_f32 + v_rcp_f32 (no ocml branches).
static __device__ __forceinline__ float fast_sigmoid(float x) {
  return __builtin_amdgcn_rcpf(1.0f + __expf(-x));
}
static __device__ __forceinline__ float fast_tanh(float x) {
#if __has_builtin(__builtin_amdgcn_tanhf)
  return __builtin_amdgcn_tanhf(x);
#else
  return 2.0f * __builtin_amdgcn_rcpf(1.0f + __expf(-2.0f * x)) - 1.0f;
#endif
}

// ---------------------------------------------------------------------------
// Weight pre-swizzle: pack Wf/Wg into per-lane WMMA A-fragment order so the
// layer kernel loads each fragment with one coalesced b64.
//   wfT[((layer*8 + k)*32 + lane)*2 + {0,1}],  k = tap*4 + kc
// ---------------------------------------------------------------------------
__global__ void __launch_bounds__(256)
wn_prep(const float* __restrict__ wf, const float* __restrict__ wg,
        float* __restrict__ wfT, float* __restrict__ wgT) {
  int idx = blockIdx.x * blockDim.x + threadIdx.x;   // over (L-1)*8*32
  if (idx >= (WN_L - 1) * 8 * 32) return;
  int lane = idx & 31;
  int k = (idx >> 5) & 7;
  int layer = idx >> 8;
  int n = lane & 15, hi = lane >> 4;
  int tap = k >> 2, kc = k & 3;
  int c0 = 4 * kc + 2 * hi;
  const float* wfl = wf + (size_t)layer * WN_C * WN_C * 2;
  const float* wgl = wg + (size_t)layer * WN_C * WN_C * 2;
  wfT[idx * 2 + 0] = wfl[(n * WN_C + c0) * 2 + tap];
  wfT[idx * 2 + 1] = wfl[(n * WN_C + c0 + 1) * 2 + tap];
  wgT[idx * 2 + 0] = wgl[(n * WN_C + c0) * 2 + tap];
  wgT[idx * 2 + 1] = wgl[(n * WN_C + c0 + 1) * 2 + tap];
}

// ---------------------------------------------------------------------------
// Layer 0: xin = x/32768;  out0 = causal conv1x2 (1->16);  agg = out0;
//          h = xin + out0  (broadcast residual).  Pair-interleaved stores.
// ---------------------------------------------------------------------------
__global__ void __launch_bounds__(256)
wn_layer0(const float* __restrict__ x, const float* __restrict__ w0,
          const float* __restrict__ b0, float* __restrict__ h,
          float* __restrict__ agg) {
  int idx = blockIdx.x * blockDim.x + threadIdx.x;   // over B*N
  int b = idx >> 15;                                 // N = 32768
  int n = idx & (WN_N - 1);
  const float inv = 1.0f / 32768.0f;
  float xc = x[idx] * inv;
  float xp = (n > 0) ? x[idx - 1] * inv : 0.0f;
  float* hb = h + (size_t)b * WN_C * WN_N;
  float* ab = agg + (size_t)b * WN_C * WN_N;
#pragma unroll
  for (int p = 0; p < 8; ++p) {
    int c0 = 2 * p, c1 = 2 * p + 1;
    float o0 = w0[c0 * 2 + 0] * xp + w0[c0 * 2 + 1] * xc + b0[c0];
    float o1 = w0[c1 * 2 + 0] * xp + w0[c1 * 2 + 1] * xc + b0[c1];
    size_t base = (size_t)p * WN_P2N + (size_t)n * 2;
    v2f a; a.x = o0; a.y = o1;
    v2f hh; hh.x = xc + o0; hh.y = xc + o1;
    *(v2f*)&ab[base] = a;
    *(v2f*)&hb[base] = hh;
  }
}

// ---------------------------------------------------------------------------
// Gated residual layer body: 4 consecutive 16-time tiles, 16 WMMAs each.
// GUARD=false is the hot path (whole group strictly after the dilation
// boundary -> unconditional delayed-tap b64 loads, zero divergence).
// ---------------------------------------------------------------------------
template <bool GUARD>
static __device__ __forceinline__ void
wn_layer_tiles(const float* __restrict__ hb, float* __restrict__ ab,
               float* __restrict__ ho, const v2f* Af, const v2f* Ag,
               const float* __restrict__ bfv, const float* __restrict__ bgv,
               int t0, int n, int hi, int d) {
#pragma unroll
  for (int tt = 0; tt < WN_TPW; ++tt) {
    int tc = t0 + tt * 16 + n;   // current-tap time
    int td = tc - d;             // delayed-tap time

    // B fragments (shared by f and g): [0..3]=tap0 (t-d), [4..7]=tap1 (t).
    // Each is one b64 from the pair-interleaved h.
    v2f Bf[8];
#pragma unroll
    for (int kc = 0; kc < 4; ++kc) {
      int p0 = 2 * kc + hi;      // pair index of channels {4kc+2hi, +1}
      size_t pbase = (size_t)p0 * WN_P2N;
      v2f bd = {};
      if (GUARD) {
        if (td >= 0) bd = *(const v2f*)&hb[pbase + (size_t)td * 2];
      } else {
        bd = *(const v2f*)&hb[pbase + (size_t)td * 2];
      }
      Bf[kc] = bd;
      Bf[4 + kc] = *(const v2f*)&hb[pbase + (size_t)tc * 2];
    }

    v8f cf = {}; v8f cg = {};
#pragma unroll
    for (int k = 0; k < 8; ++k) {
      cf = WMMA_F32X4(Af[k], Bf[k], cf);
      cg = WMMA_F32X4(Ag[k], Bf[k], cg);
    }

    // Epilogue: channels r,r+1 share a pair -> b64 RMW for agg and h.
#pragma unroll
    for (int rp = 0; rp < 4; ++rp) {
      int p = 4 * hi + rp;       // pair of channels {8hi+2rp, +1}
      size_t base = (size_t)p * WN_P2N + (size_t)tc * 2;
      float f0 = cf[2 * rp + 0] + bfv[2 * rp + 0];
      float g0 = cg[2 * rp + 0] + bgv[2 * rp + 0];
      float f1 = cf[2 * rp + 1] + bfv[2 * rp + 1];
      float g1 = cg[2 * rp + 1] + bgv[2 * rp + 1];
      float o0 = fast_tanh(f0) * fast_sigmoid(g0);
      float o1 = fast_tanh(f1) * fast_sigmoid(g1);
      v2f aold = *(const v2f*)&ab[base];
      v2f hold = *(const v2f*)&hb[base];
      v2f an; an.x = aold.x + o0; an.y = aold.y + o1;
      v2f hn; hn.x = hold.x + o0; hn.y = hold.y + o1;
      *(v2f*)&ab[base] = an;
      *(v2f*)&ho[base] = hn;
    }
  }
}

__global__ void __launch_bounds__(256)
wn_layer(const float* __restrict__ hin, float* __restrict__ hout,
         float* __restrict__ agg,
         const float* __restrict__ wfT, const float* __restrict__ bf,
         const float* __restrict__ wgT, const float* __restrict__ bg, int d) {
  int wave = blockIdx.x * (blockDim.x >> 5) + (threadIdx.x >> 5);
  int lane = threadIdx.x & 31;
  int n = lane & 15;           // tile column (time) / A-row
  int hi = lane >> 4;          // which half of the wave
  const int groupsPerB = (WN_N / 16) / WN_TPW;       // 512
  int b = wave / groupsPerB;
  int t0 = (wave % groupsPerB) * (16 * WN_TPW);

  const float* hb = hin + (size_t)b * WN_C * WN_N;
  float* ab = agg + (size_t)b * WN_C * WN_N;
  float* ho = hout + (size_t)b * WN_C * WN_N;

  // A fragments: one coalesced b64 per fragment from the pre-swizzled banks.
  v2f Af[8], Ag[8];
#pragma unroll
  for (int k = 0; k < 8; ++k) {
    Af[k] = *(const v2f*)&wfT[(k * 32 + lane) * 2];
    Ag[k] = *(const v2f*)&wgT[(k * 32 + lane) * 2];
  }

  // Biases for this lane's 8 channels: two b128 loads each.
  float bfv[8], bgv[8];
  v4f bl = *(const v4f*)&bf[8 * hi], bh = *(const v4f*)&bf[8 * hi + 4];
  v4f gl = *(const v4f*)&bg[8 * hi], gh = *(const v4f*)&bg[8 * hi + 4];
#pragma unroll
  for (int r = 0; r < 4; ++r) {
    bfv[r] = bl[r]; bfv[r + 4] = bh[r];
    bgv[r] = gl[r]; bgv[r + 4] = gh[r];
  }

  if (t0 >= d) {   // wave-uniform: whole 64-step group is past the boundary
    wn_layer_tiles<false>(hb, ab, ho, Af, Ag, bfv, bgv, t0, n, hi, d);
  } else {
    wn_layer_tiles<true>(hb, ab, ho, Af, Ag, bfv, bgv, t0, n, hi, d);
  }
}

// ---------------------------------------------------------------------------
// Head: a = Wa(64x16)·relu(agg)+ba; o = Wo(256x64)·relu(a)+bo;
//       log_softmax over 256 channels; mask by lengths; write (B,256,N).
// One wave per (b, 16-time tile). Intermediate 64x16 relu(a) tile goes
// through per-wave LDS (stride 17 to dodge bank conflicts) to re-fragment
// it as the B operand of the second matmul. 16+256 WMMAs per tile.
// ---------------------------------------------------------------------------
__global__ void __launch_bounds__(128)
wn_head(const float* __restrict__ agg, const int* __restrict__ lengths,
        const float* __restrict__ wa, const float* __restrict__ ba,
        const float* __restrict__ wo, const float* __restrict__ bo,
        float* __restrict__ out) {
  __shared__ float alds[4][64 * 17];
  int wv = threadIdx.x >> 5;
  int wave = blockIdx.x * (blockDim.x >> 5) + wv;
  int lane = threadIdx.x & 31;
  int n = lane & 15;
  int hi = lane >> 4;
  const int tilesPerB = WN_N >> 4;
  int b = wave / tilesPerB;
  int t0 = (wave % tilesPerB) << 4;
  int tc = t0 + n;
  const float* ab = agg + (size_t)b * WN_C * WN_N;
  int m = n;
  float* L = alds[wv];

  // B fragments of relu(agg): 16 K-rows -> 4 b64 loads (pair-interleaved).
  v2f B1[4];
#pragma unroll
  for (int kc = 0; kc < 4; ++kc) {
    int p0 = 2 * kc + hi;
    v2f raw = *(const v2f*)&ab[(size_t)p0 * WN_P2N + (size_t)tc * 2];
    v2f bb;
    bb.x = fmaxf(raw.x, 0.0f);
    bb.y = fmaxf(raw.y, 0.0f);
    B1[kc] = bb;
  }

  // a = Wa·relu(agg) + ba, relu, deposit 64x16 tile to LDS.
#pragma unroll
  for (int mt = 0; mt < 4; ++mt) {
    v8f acc = {};
#pragma unroll
    for (int kc = 0; kc < 4; ++kc) {
      int c0 = 4 * kc + 2 * hi;
      v2f a = *(const v2f*)&wa[(mt * 16 + m) * 16 + c0];  // contiguous pair
      acc = WMMA_F32X4(a, B1[kc], acc);
    }
    v4f bal = *(const v4f*)&ba[mt * 16 + 8 * hi];
    v4f bah = *(const v4f*)&ba[mt * 16 + 8 * hi + 4];
#pragma unroll
    for (int r = 0; r < 8; ++r) {
      int row = mt * 16 + r + 8 * hi;
      float bia = (r < 4) ? bal[r & 3] : bah[r & 3];
      L[row * 17 + n] = fmaxf(acc[r] + bia, 0.0f);
    }
  }
  // Per-wave LDS region; DS ops are in-order within a wave -> no barrier.

  // Re-fragment relu(a) as B operand: 64 K-rows -> 16 chunks.
  v2f B2[16];
#pragma unroll
  for (int kc = 0; kc < 16; ++kc) {
    int r0 = 4 * kc + 2 * hi;
    v2f bb;
    bb.x = L[r0 * 17 + n];
    bb.y = L[(r0 + 1) * 17 + n];
    B2[kc] = bb;
  }

  // o = Wo·relu(a): 16 row-tiles x 16 K-chunks = 256 WMMAs; keep all 256
  // logits for this column in 16 v8f accumulators.
  v8f co[16];
#pragma unroll
  for (int mt2 = 0; mt2 < 16; ++mt2) {
    v8f acc = {};
#pragma unroll
    for (int kc = 0; kc < 16; ++kc) {
      int c0 = 4 * kc + 2 * hi;
      v2f a = *(const v2f*)&wo[(mt2 * 16 + m) * 64 + c0];  // contiguous pair
      acc = WMMA_F32X4(a, B2[kc], acc);
    }
    co[mt2] = acc;
  }

  // + bo, then log-softmax over 256 channels. Each lane holds 128 logits of
  // its column; the other 128 live in lane^16 -> one shfl_xor per reduction.
  float mx = -3.402823466e38f;
#pragma unroll
  for (int mt2 = 0; mt2 < 16; ++mt2) {
    v4f bol = *(const v4f*)&bo[mt2 * 16 + 8 * hi];
    v4f boh = *(const v4f*)&bo[mt2 * 16 + 8 * hi + 4];
#pragma unroll
    for (int r = 0; r < 8; ++r) {
      float bia = (r < 4) ? bol[r & 3] : boh[r & 3];
      float v = co[mt2][r] + bia;
      co[mt2][r] = v;
      mx = fmaxf(mx, v);
    }
  }
  mx = fmaxf(mx, __shfl_xor(mx, 16, 32));
  float s = 0.0f;
#pragma unroll
  for (int mt2 = 0; mt2 < 16; ++mt2) {
#pragma unroll
    for (int r = 0; r < 8; ++r) s += __expf(co[mt2][r] - mx);
  }
  s += __shfl_xor(s, 16, 32);
  float lz = __logf(s) + mx;

  float msk = (tc < lengths[b]) ? 1.0f : 0.0f;
  float* ob = out + (size_t)b * 256 * WN_N + tc;
#pragma unroll
  for (int mt2 = 0; mt2 < 16; ++mt2) {
#pragma unroll
    for (int r = 0; r < 8; ++r) {
      int row = mt2 * 16 + r + 8 * hi;
      ob[(size_t)row * WN_N] = (co[mt2][r] - lz) * msk;
    }
  }
}

// ---------------------------------------------------------------------------
extern "C" void kernel_launch(void* const* d_in, const int* in_sizes, int n_in,
                              void* d_out, int out_size, void* d_ws, size_t ws_size,
                              hipStream_t stream) {
  const float* x  = (const float*)d_in[0];
  const int* lens = (const int*)d_in[1];
  const float* w0 = (const float*)d_in[2];
  const float* b0 = (const float*)d_in[3];
  const float* wf = (const float*)d_in[4];
  const float* bf = (const float*)d_in[5];
  const float* wg = (const float*)d_in[6];
  const float* bg = (const float*)d_in[7];
  const float* wa = (const float*)d_in[8];
  const float* ba = (const float*)d_in[9];
  const float* wo = (const float*)d_in[10];
  const float* bo = (const float*)d_in[11];
  float* out = (float*)d_out;

  const size_t hElems = (size_t)WN_B * WN_C * WN_N;   // 4M floats = 16 MB
  float* hA  = (float*)d_ws;
  float* hB  = hA + hElems;
  float* agg = hB + hElems;
  float* wfT = agg + hElems;                          // 29*512 floats = 58 KB
  float* wgT = wfT + (size_t)(WN_L - 1) * WN_FRAG_PER_LAYER;

  const int prepThreads = (WN_L - 1) * 8 * 32;        // 7424
  wn_prep<<<(prepThreads + 255) / 256, 256, 0, stream>>>(wf, wg, wfT, wgT);
  wn_layer0<<<(WN_B * WN_N) / 256, 256, 0, stream>>>(x, w0, b0, hA, agg);

  const int tileWaves = WN_B * (WN_N / 16);           // 16384 tiles/layer
  const int layerWaves = tileWaves / WN_TPW;          // 4096 waves/layer
  for (int i = 1; i < WN_L; ++i) {
    int d = 1 << (i % 10);
    const float* hin = (i & 1) ? hA : hB;
    float* hout      = (i & 1) ? hB : hA;
    wn_layer<<<layerWaves / 8, 256, 0, stream>>>(
        hin, hout, agg,
        wfT + (size_t)(i - 1) * WN_FRAG_PER_LAYER, bf + (size_t)(i - 1) * WN_C,
        wgT + (size_t)(i - 1) * WN_FRAG_PER_LAYER, bg + (size_t)(i - 1) * WN_C, d);
  }

  wn_head<<<tileWaves / 4, 128, 0, stream>>>(agg, lens, wa, ba, wo, bo, out);
}